// Pointnet2Backbone_78048145703224
// MI455X (gfx1250) — compile-verified
//
#include <hip/hip_runtime.h>
#include <hip/hip_bf16.h>
#include <stdint.h>

typedef float v2f __attribute__((ext_vector_type(2)));
typedef float v8f __attribute__((ext_vector_type(8)));

#define B_ 16
#define N_ 2048
#define C_ 256
#define K_ 16
#define OUTC_ (3 + C_)

// ---------------------------------------------------------------------------
// Kernel 1: kNN via WMMA f32 16x16x4 Gram matrix + per-row top-16 selection.
// One wave32 per (batch, 16-row tile).  grid = B * (N/16) = 2048 blocks.
// ---------------------------------------------------------------------------
__global__ void __launch_bounds__(32) knn_kernel(const float* __restrict__ xyz,
                                                 int* __restrict__ knnIdx) {
  __shared__ __align__(16) float xyzS[N_ * 3];   // 24 KB: whole batch xyz
  __shared__ float tileD[16 * 35];               // two 16x16 d2 tiles (cols 0,17)

  const int b       = blockIdx.x >> 7;           // N_/16 = 128 row tiles/batch
  const int rowBase = (blockIdx.x & 127) << 4;
  const int lane    = threadIdx.x;
  const bool hi     = lane >= 16;
  const int  nl     = lane & 15;

  // Stage this batch's xyz into LDS with gfx1250 async global->LDS copies
  // (tracked by ASYNCcnt, no VGPR round trip).
  {
    const float4* src4 = (const float4*)(xyz + (size_t)b * N_ * 3);
    for (int i = lane; i < (N_ * 3) / 4; i += 32) {
      uint32_t ldsoff = (uint32_t)(uintptr_t)(&xyzS[i * 4]);
      const float4* g = src4 + i;
      asm volatile("global_load_async_to_lds_b128 %0, %1, off"
                   :: "v"(ldsoff), "v"(g) : "memory");
    }
    asm volatile("s_wait_asynccnt 0" ::: "memory");
  }
  __syncthreads();

  // A operand: 16x4 fp32 tile of the 16 rows (K=3 padded with 0).
  // lanes 0-15: (K0,K1)=(x,y) of row M=lane; lanes 16-31: (K2,K3)=(z,0).
  const int arow = rowBase + nl;
  const float ax = xyzS[arow * 3 + 0];
  const float ay = xyzS[arow * 3 + 1];
  const float az = xyzS[arow * 3 + 2];
  v2f avec;
  avec.x = hi ? az : ax;
  avec.y = hi ? 0.0f : ay;

  // |x_m|^2 for the 8 accumulator rows this lane owns in the C/D layout.
  float sqRow[8];
#pragma unroll
  for (int r = 0; r < 8; ++r) {
    const int m = rowBase + r + (hi ? 8 : 0);
    const float x = xyzS[m * 3 + 0];
    const float y = xyzS[m * 3 + 1];
    const float z = xyzS[m * 3 + 2];
    sqRow[r] = x * x + y * y + z * z;
  }

  // Register-resident sorted top-16 (ascending), self excluded.
  float bestD[K_];
  int   bestI[K_];
#pragma unroll
  for (int t = 0; t < K_; ++t) { bestD[t] = 3.4e38f; bestI[t] = 0; }
  const int gRow = rowBase + nl;   // the row lanes 0-15 select for

  for (int ct = 0; ct < N_ / 16; ct += 2) {
    const int colBase = ct << 4;

    // Two B operands (two adjacent 16-column tiles) -> two back-to-back WMMAs.
    const int n0 = colBase + nl;
    const int n1 = n0 + 16;
    const float b0x = xyzS[n0 * 3 + 0], b0y = xyzS[n0 * 3 + 1], b0z = xyzS[n0 * 3 + 2];
    const float b1x = xyzS[n1 * 3 + 0], b1y = xyzS[n1 * 3 + 1], b1z = xyzS[n1 * 3 + 2];
    const float sqn0 = b0x * b0x + b0y * b0y + b0z * b0z;
    const float sqn1 = b1x * b1x + b1y * b1y + b1z * b1z;
    v2f bvec0, bvec1;
    bvec0.x = hi ? b0z : b0x;  bvec0.y = hi ? 0.0f : b0y;
    bvec1.x = hi ? b1z : b1x;  bvec1.y = hi ? 0.0f : b1y;

    v8f acc0 = {};
    v8f acc1 = {};
    acc0 = __builtin_amdgcn_wmma_f32_16x16x4_f32(false, avec, false, bvec0,
                                                 (short)0, acc0, false, false);
    acc1 = __builtin_amdgcn_wmma_f32_16x16x4_f32(false, avec, false, bvec1,
                                                 (short)0, acc1, false, false);

    // d2 = |x_m|^2 + |x_n|^2 - 2 dot  -> LDS, row stride 35, tiles at col 0/17
#pragma unroll
    for (int r = 0; r < 8; ++r) {
      const int rr = r + (hi ? 8 : 0);
      tileD[rr * 35 + nl]      = sqRow[r] + sqn0 - 2.0f * acc0[r];
      tileD[rr * 35 + 17 + nl] = sqRow[r] + sqn1 - 2.0f * acc1[r];
    }
    __syncthreads();

    // Lanes 0-15: scan 32 candidates; tight non-unrolled loop with a rare
    // branchless bubble-insert (guard fails almost always after warm-up).
    if (lane < 16) {
#pragma unroll 1
      for (int j = 0; j < 32; ++j) {
        const int m = colBase + j;
        const float d = tileD[lane * 35 + j + (j >> 4)];
        if (m != gRow && d < bestD[K_ - 1]) {
          float dd = d; int ii = m;
#pragma unroll
          for (int t = 0; t < K_; ++t) {
            const float curD = bestD[t];
            const int   curI = bestI[t];
            const bool  sm   = dd < curD;
            bestD[t] = sm ? dd : curD;
            bestI[t] = sm ? ii : curI;
            dd = sm ? curD : dd;
            ii = sm ? curI : ii;
          }
        }
      }
    }
    __syncthreads();
  }

  if (lane < 16) {
    const int base = (b * N_ + gRow) * K_;
#pragma unroll
    for (int t = 0; t < K_; ++t) knnIdx[base + t] = bestI[t];
  }
}

// ---------------------------------------------------------------------------
// Kernel 2: transpose features (B,C,N) -> featT (B,N,C) so neighbor gathers
// become contiguous 1KB rows (coalesced, L2-resident).
// ---------------------------------------------------------------------------
__global__ void transpose_kernel(const float* __restrict__ features,
                                 float* __restrict__ featT) {
  __shared__ float tile[32][33];
  const int b  = blockIdx.z;
  const int nT = blockIdx.x * 32;
  const int cT = blockIdx.y * 32;
  const int tx = threadIdx.x, ty = threadIdx.y;

  const float* src = features + (size_t)b * C_ * N_;
#pragma unroll
  for (int yy = 0; yy < 4; ++yy) {
    const int cc = cT + ty + yy * 8;
    tile[ty + yy * 8][tx] = src[(size_t)cc * N_ + nT + tx];   // coalesced in n
  }
  __syncthreads();
  float* dst = featT + (size_t)b * N_ * C_;
#pragma unroll
  for (int yy = 0; yy < 4; ++yy) {
    const int nn = nT + ty + yy * 8;
    dst[(size_t)nn * C_ + cT + tx] = tile[tx][ty + yy * 8];   // coalesced in c
  }
}

// ---------------------------------------------------------------------------
// Kernel 3: gather + mean.  One block (256 threads) per point; thread c owns
// channel c.  out[bn][0:3] = mean(nbr xyz) - xyz;  out[bn][3+c] = mean - self.
// ---------------------------------------------------------------------------
__global__ void __launch_bounds__(C_) gather_kernel(const float* __restrict__ xyz,
                                                    const float* __restrict__ featT,
                                                    const int* __restrict__ knnIdx,
                                                    float* __restrict__ out) {
  const int bn = blockIdx.x;               // 0 .. B*N-1
  const int b  = bn >> 11;                 // N_ = 2048
  __shared__ int idxS[K_];
  const int t = threadIdx.x;
  if (t < K_) idxS[t] = knnIdx[bn * K_ + t];
  __syncthreads();

  float sum = 0.0f;
#pragma unroll
  for (int j = 0; j < K_; ++j) {
    const float* row = featT + (size_t)(b * N_ + idxS[j]) * C_;
    if (j + 1 < K_)
      __builtin_prefetch(featT + (size_t)(b * N_ + idxS[j + 1]) * C_ + t, 0, 1);
    sum += row[t];                         // contiguous 1KB per neighbor
  }
  const float self = featT[(size_t)bn * C_ + t];
  out[(size_t)bn * OUTC_ + 3 + t] = sum * (1.0f / K_) - self;

  if (t < 3) {
    float s = 0.0f;
#pragma unroll
    for (int j = 0; j < K_; ++j) s += xyz[(size_t)(b * N_ + idxS[j]) * 3 + t];
    out[(size_t)bn * OUTC_ + t] = s * (1.0f / K_) - xyz[(size_t)bn * 3 + t];
  }
}

// ---------------------------------------------------------------------------
extern "C" void kernel_launch(void* const* d_in, const int* in_sizes, int n_in,
                              void* d_out, int out_size, void* d_ws, size_t ws_size,
                              hipStream_t stream) {
  const float* xyz      = (const float*)d_in[0];   // (B,N,3) f32
  const float* features = (const float*)d_in[1];   // (B,C,N) f32
  float* out = (float*)d_out;                      // (B,N,3+C) f32

  // workspace: [featT: B*N*C f32][knnIdx: B*N*K i32]
  float* featT = (float*)d_ws;
  int*   knnIdx = (int*)((char*)d_ws + (size_t)B_ * N_ * C_ * sizeof(float));

  knn_kernel<<<B_ * (N_ / 16), 32, 0, stream>>>(xyz, knnIdx);
  transpose_kernel<<<dim3(N_ / 32, C_ / 32, B_), dim3(32, 8), 0, stream>>>(features, featT);
  gather_kernel<<<B_ * N_, C_, 0, stream>>>(xyz, featT, knnIdx, out);
}